// GPT2Decode_46119358824836
// MI455X (gfx1250) — compile-verified
//
#include <hip/hip_runtime.h>
#include <hip/hip_bf16.h>
#include <math.h>

// ---------------------------------------------------------------------------
// GPT-2 single-token decode for gfx1250 (MI455X).
// Everything is memory-bandwidth-bound GEMV; fp32 throughout (matches ref).
// GEMVs are routed through V_WMMA_F32_16X16X4_F32 (exact fp32 MAC path).
// Activations are staged in LDS so the global-load stream is purely weights.
// ---------------------------------------------------------------------------

#define L_ 12
#define H_ 12
#define D_ 64
#define E_ 768
#define V_ 50257
#define MS_ 1024

typedef __attribute__((ext_vector_type(2))) float v2f;
typedef __attribute__((ext_vector_type(8))) float v8f;

// ---------------------------------------------------------------------------
// Embedding: h = wte[id] + wpe[S]
// ---------------------------------------------------------------------------
__global__ void embed_kernel(const int* __restrict__ ids,
                             const float* __restrict__ wte,
                             const float* __restrict__ wpe,
                             const int* __restrict__ pS,
                             float* __restrict__ h) {
    const int S  = *pS;
    const int id = ids[0];
    for (int e = threadIdx.x; e < E_; e += blockDim.x)
        h[e] = wte[(size_t)id * E_ + e] + wpe[(size_t)S * E_ + e];
}

// ---------------------------------------------------------------------------
// LayerNorm over E=768, one block of 256 threads (3 elems/thread)
// ---------------------------------------------------------------------------
__global__ void ln_kernel(const float* __restrict__ in,
                          const float* __restrict__ g,
                          const float* __restrict__ b,
                          float* __restrict__ out) {
    __shared__ float red[256];
    __shared__ float s_mean, s_rstd;
    const int t = threadIdx.x;

    float v0 = in[t], v1 = in[t + 256], v2 = in[t + 512];
    red[t] = v0 + v1 + v2;
    __syncthreads();
    for (int s = 128; s > 0; s >>= 1) {
        if (t < s) red[t] += red[t + s];
        __syncthreads();
    }
    if (t == 0) s_mean = red[0] * (1.0f / E_);
    __syncthreads();
    const float m = s_mean;
    float d0 = v0 - m, d1 = v1 - m, d2 = v2 - m;
    red[t] = d0 * d0 + d1 * d1 + d2 * d2;
    __syncthreads();
    for (int s = 128; s > 0; s >>= 1) {
        if (t < s) red[t] += red[t + s];
        __syncthreads();
    }
    if (t == 0) s_rstd = rsqrtf(red[0] * (1.0f / E_) + 1e-5f);
    __syncthreads();
    const float r = s_rstd;
    out[t]       = d0 * r * g[t]       + b[t];
    out[t + 256] = d1 * r * g[t + 256] + b[t + 256];
    out[t + 512] = d2 * r * g[t + 512] + b[t + 512];
}

// ---------------------------------------------------------------------------
// WMMA-based GEMV:  out[n] = act( x . W[:,n] + bias[n] ) (+ resid[n])
//   TRANSB=0 : W row-major [K][N]   (n contiguous)      -> weight matrices
//   TRANSB=1 : W row-major [N][K]   (k contiguous)      -> wte for LM head
// One wave computes 16 outputs. A-tile rows are all x[k..k+3], so every row
// of the D accumulator equals the GEMV result; we read row 0 (lanes 0..15).
// x is staged in dynamic LDS: A fetch = one ds_load_b64 (broadcast), so the
// VMEM stream is weights only.
//
// f32 A 16x4 layout:  lanes 0-15: VGPR0=K0, VGPR1=K1; lanes 16-31: K2,K3
// f32 B 4x16 layout:  lanes 0-15 hold N=0..15; VGPR0/1 = rows K0/K1 (lanes
//                     0-15) and K2/K3 (lanes 16-31).
// ---------------------------------------------------------------------------
#define ACT_NONE 0
#define ACT_GELU 1

template <int ACT, bool TRANSB>
__global__ void gemv_wmma_kernel(const float* __restrict__ x,
                                 const float* __restrict__ W,
                                 const float* __restrict__ bias,
                                 const float* resid,
                                 float* out, int K, int N) {
    extern __shared__ float xlds[];                 // K floats
    // Cooperative stage of the activation vector (before any wave exits!)
    for (int i = threadIdx.x; i < K; i += blockDim.x) xlds[i] = x[i];
    __syncthreads();

    const int lane = threadIdx.x & 31;
    const int wid  = (blockIdx.x * blockDim.x + threadIdx.x) >> 5; // uniform per wave
    const int n0   = wid << 4;
    if (n0 >= N) return;                       // wave-uniform exit (EXEC all-1 for WMMA)

    const int colOff = lane & 15;
    const int col    = n0 + colOff;
    const int cc     = (col < N) ? col : (N - 1);   // clamp for edge tile loads
    const int khalf  = (lane >> 4) << 1;            // 0 (lanes 0-15) or 2 (16-31)

    v8f acc = {};
    if (TRANSB) {
        const float* base = W + (size_t)cc * K + khalf;
        for (int k = 0; k < K; k += 4) {
            v2f bv = *reinterpret_cast<const v2f*>(base + k);            // global_load_b64
            v2f av = *reinterpret_cast<const v2f*>(xlds + k + khalf);    // ds_load_b64
            acc = __builtin_amdgcn_wmma_f32_16x16x4_f32(
                false, av, false, bv, (short)0, acc, false, false);
        }
    } else {
        const float* base = W + cc;
        for (int k = 0; k < K; k += 4) {
            v2f bv;
            bv.x = base[(size_t)(k + khalf)     * N];
            bv.y = base[(size_t)(k + khalf + 1) * N];
            v2f av = *reinterpret_cast<const v2f*>(xlds + k + khalf);    // ds_load_b64
            acc = __builtin_amdgcn_wmma_f32_16x16x4_f32(
                false, av, false, bv, (short)0, acc, false, false);
        }
    }

    float r = acc[0];                      // D row 0, N = lane (lanes 0-15)
    if (lane < 16 && col < N) {
        if (bias) r += bias[col];
        if (ACT == ACT_GELU) {
            float u = r;
            r = 0.5f * u * (1.0f + tanhf(0.7978845608028654f * (u + 0.044715f * u * u * u)));
        }
        if (resid) r += resid[col];
        out[col] = r;
    }
}

// ---------------------------------------------------------------------------
// Write fresh k,v (from qkv buffer) into the output KV cache at row S.
// ---------------------------------------------------------------------------
__global__ void kv_write_kernel(const float* __restrict__ qkv,
                                const int* __restrict__ pS,
                                float* __restrict__ opk,
                                float* __restrict__ opv,
                                int layer) {
    const int S = *pS;
    for (int i = threadIdx.x; i < E_; i += blockDim.x) {
        const int h = i >> 6, d = i & 63;
        const size_t idx = ((((size_t)layer * H_) + h) * MS_ + S) * D_ + d;
        opk[idx] = qkv[E_ + i];        // k slice
        opv[idx] = qkv[2 * E_ + i];    // v slice
    }
}

// ---------------------------------------------------------------------------
// Attention: one block per head. scores -> softmax -> o. Reads the *output*
// KV cache (already patched with row S), so k_full/v_full are contiguous.
// ---------------------------------------------------------------------------
__global__ void attn_kernel(const float* __restrict__ qkv,
                            const float* __restrict__ pk,
                            const float* __restrict__ pv,
                            const float* __restrict__ mask,
                            const int* __restrict__ pS,
                            float* __restrict__ o,
                            int layer) {
    __shared__ float qs[D_];
    __shared__ float sc[MS_];
    __shared__ float red[256];
    __shared__ float part[4][D_];
    __shared__ float s_max, s_sum;

    const int t    = threadIdx.x;
    const int head = blockIdx.x;
    const int S1   = *pS + 1;                       // number of valid keys

    if (t < D_) qs[t] = qkv[head * D_ + t];
    __syncthreads();

    const size_t hb = (((size_t)layer * H_) + head) * MS_;

    // scores
    float lmax = -INFINITY;
    for (int j = t; j < MS_; j += 256) {
        float s = -INFINITY;
        if (j < S1) {
            const float* krow = pk + (hb + j) * D_;
            float acc = 0.0f;
            #pragma unroll
            for (int d4 = 0; d4 < D_ / 4; ++d4) {
                float4 kk = reinterpret_cast<const float4*>(krow)[d4];
                acc += qs[4 * d4 + 0] * kk.x + qs[4 * d4 + 1] * kk.y
                     + qs[4 * d4 + 2] * kk.z + qs[4 * d4 + 3] * kk.w;
            }
            s = acc * 0.125f + (1.0f - mask[j]) * (-1e9f);
        }
        sc[j] = s;
        lmax = fmaxf(lmax, s);
    }
    red[t] = lmax;
    __syncthreads();
    for (int s = 128; s > 0; s >>= 1) {
        if (t < s) red[t] = fmaxf(red[t], red[t + s]);
        __syncthreads();
    }
    if (t == 0) s_max = red[0];
    __syncthreads();
    const float mx = s_max;

    // exp + sum
    float lsum = 0.0f;
    for (int j = t; j < MS_; j += 256) {
        float p = (j < S1) ? __expf(sc[j] - mx) : 0.0f;
        sc[j] = p;
        lsum += p;
    }
    __syncthreads();
    red[t] = lsum;
    __syncthreads();
    for (int s = 128; s > 0; s >>= 1) {
        if (t < s) red[t] += red[t + s];
        __syncthreads();
    }
    if (t == 0) s_sum = red[0];
    __syncthreads();
    const float inv = 1.0f / s_sum;

    // o[d] = sum_j p[j] * V[j][d]
    const int d   = t & 63;
    const int grp = t >> 6;
    float acc = 0.0f;
    for (int j = grp; j < MS_; j += 4)
        acc += sc[j] * pv[(hb + j) * D_ + d];
    part[grp][d] = acc;
    __syncthreads();
    if (t < D_)
        o[head * D_ + t] = (part[0][t] + part[1][t] + part[2][t] + part[3][t]) * inv;
}

// ---------------------------------------------------------------------------
// Host-side launch
// ---------------------------------------------------------------------------
extern "C" void kernel_launch(void* const* d_in, const int* in_sizes, int n_in,
                              void* d_out, int out_size, void* d_ws, size_t ws_size,
                              hipStream_t stream) {
    (void)in_sizes; (void)n_in; (void)out_size; (void)ws_size;

    const int*   ids     = (const int*)  d_in[0];
    const float* past_k  = (const float*)d_in[1];
    const float* past_v  = (const float*)d_in[2];
    const int*   pS      = (const int*)  d_in[3];
    const float* amask   = (const float*)d_in[4];
    const float* wte     = (const float*)d_in[5];
    const float* wpe     = (const float*)d_in[6];
    const float* ln1_g   = (const float*)d_in[7];
    const float* ln1_b   = (const float*)d_in[8];
    const float* attn_w  = (const float*)d_in[9];
    const float* attn_b  = (const float*)d_in[10];
    const float* proj_w  = (const float*)d_in[11];
    const float* proj_b  = (const float*)d_in[12];
    const float* ln2_g   = (const float*)d_in[13];
    const float* ln2_b   = (const float*)d_in[14];
    const float* fc_w    = (const float*)d_in[15];
    const float* fc_b    = (const float*)d_in[16];
    const float* mlp_w   = (const float*)d_in[17];
    const float* mlp_b   = (const float*)d_in[18];
    const float* lnf_g   = (const float*)d_in[19];
    const float* lnf_b   = (const float*)d_in[20];

    float* logits = (float*)d_out;
    const size_t kvElems = (size_t)L_ * H_ * MS_ * D_;
    float* opk = logits + V_;
    float* opv = opk + kvElems;

    // workspace layout
    float* h    = (float*)d_ws;          // 768
    float* x    = h + E_;                // 768
    float* qkv  = x + E_;                // 2304
    float* ovec = qkv + 3 * E_;          // 768
    float* fcb  = ovec + E_;             // 3072

    // output KV cache = input cache (row S patched per layer below)
    hipMemcpyAsync(opk, past_k, kvElems * sizeof(float), hipMemcpyDeviceToDevice, stream);
    hipMemcpyAsync(opv, past_v, kvElems * sizeof(float), hipMemcpyDeviceToDevice, stream);

    embed_kernel<<<1, 256, 0, stream>>>(ids, wte, wpe, pS, h);

    auto blocksFor = [](int N) { int waves = (N + 15) / 16; return (waves + 7) / 8; };

    for (int l = 0; l < L_; ++l) {
        ln_kernel<<<1, 256, 0, stream>>>(h, ln1_g + (size_t)l * E_, ln1_b + (size_t)l * E_, x);

        // qkv = x @ attn_w[l] + attn_b[l]           (768 x 2304)
        gemv_wmma_kernel<ACT_NONE, false>
            <<<blocksFor(3 * E_), 256, E_ * sizeof(float), stream>>>(
            x, attn_w + (size_t)l * E_ * 3 * E_, attn_b + (size_t)l * 3 * E_,
            nullptr, qkv, E_, 3 * E_);

        kv_write_kernel<<<1, 256, 0, stream>>>(qkv, pS, opk, opv, l);

        attn_kernel<<<H_, 256, 0, stream>>>(qkv, opk, opv, amask, pS, ovec, l);

        // h += o @ proj_w[l] + proj_b[l]            (768 x 768)
        gemv_wmma_kernel<ACT_NONE, false>
            <<<blocksFor(E_), 256, E_ * sizeof(float), stream>>>(
            ovec, proj_w + (size_t)l * E_ * E_, proj_b + (size_t)l * E_,
            h, h, E_, E_);

        ln_kernel<<<1, 256, 0, stream>>>(h, ln2_g + (size_t)l * E_, ln2_b + (size_t)l * E_, x);

        // fc = gelu(x @ fc_w[l] + fc_b[l])          (768 x 3072)
        gemv_wmma_kernel<ACT_GELU, false>
            <<<blocksFor(4 * E_), 256, E_ * sizeof(float), stream>>>(
            x, fc_w + (size_t)l * E_ * 4 * E_, fc_b + (size_t)l * 4 * E_,
            nullptr, fcb, E_, 4 * E_);

        // h += fc @ mlp_w[l] + mlp_b[l]             (3072 x 768)
        gemv_wmma_kernel<ACT_NONE, false>
            <<<blocksFor(E_), 256, 4 * E_ * sizeof(float), stream>>>(
            fcb, mlp_w + (size_t)l * 4 * E_ * E_, mlp_b + (size_t)l * E_,
            h, h, 4 * E_, E_);
    }

    ln_kernel<<<1, 256, 0, stream>>>(h, lnf_g, lnf_b, x);

    // logits = x @ wte^T   (wte row-major [V][E] -> TRANSB path, b64 streams)
    gemv_wmma_kernel<ACT_NONE, true>
        <<<blocksFor(V_), 256, E_ * sizeof(float), stream>>>(
        x, wte, nullptr, nullptr, logits, E_, V_);
}